// CaevlFT_39367670235990
// MI455X (gfx1250) — compile-verified
//
#include <hip/hip_runtime.h>

typedef float v8f __attribute__((ext_vector_type(8)));
typedef float v2f __attribute__((ext_vector_type(2)));

#define BATCH 256
#define CH    512
#define HW    49
#define EMBD  8192

// ---------------- zero accumulators ----------------
__global__ void zero_k(float* p, int n) {
  int i = blockIdx.x * blockDim.x + threadIdx.x;
  if (i < n) p[i] = 0.f;
}

// ---------------- transpose (B,C,HW) -> (B,HW,C) ----------------
__global__ void transpose_k(const float* __restrict__ in, float* __restrict__ out) {
  __shared__ float tile[64][50];
  int b  = blockIdx.x >> 3;
  int c0 = (blockIdx.x & 7) * 64;
  int tid = threadIdx.x;
  for (int idx = tid; idx < 64 * HW; idx += 256) {
    int ci = idx / HW, hw = idx % HW;
    tile[ci][hw] = in[((size_t)b * CH + c0 + ci) * HW + hw];
  }
  __syncthreads();
  for (int idx = tid; idx < HW * 64; idx += 256) {
    int hw = idx >> 6, ci = idx & 63;
    out[((size_t)b * HW + hw) * CH + c0 + ci] = tile[ci][hw];
  }
}

// ---------------- branch-1 matching: feature-space NN (both directions) ----------------
__global__ void match1_k(const float* __restrict__ t1, const float* __restrict__ t2,
                         int* __restrict__ idx1, int* __restrict__ idx2) {
  __shared__ float n1[HW], n2[HW];
  __shared__ float d[HW][HW];
  int b = blockIdx.x, tid = threadIdx.x;
  const float* r1 = t1 + (size_t)b * HW * CH;
  const float* r2 = t2 + (size_t)b * HW * CH;
  for (int idx = tid; idx < 2 * HW; idx += 256) {
    int m = (idx < HW) ? idx : idx - HW;
    const float* r = (idx < HW) ? (r1 + (size_t)m * CH) : (r2 + (size_t)m * CH);
    float s = 0.f;
    for (int c = 0; c < CH; ++c) { float v = r[c]; s += v * v; }
    if (idx < HW) n1[m] = s; else n2[m] = s;
  }
  __syncthreads();
  for (int pair = tid; pair < HW * HW; pair += 256) {
    int m = pair / HW, n = pair % HW;
    const float* a  = r1 + (size_t)m * CH;
    const float* c2 = r2 + (size_t)n * CH;
    float dot = 0.f;
    for (int c = 0; c < CH; ++c) dot += a[c] * c2[c];
    d[m][n] = n1[m] + n2[n] - 2.f * dot;
  }
  __syncthreads();
  if (tid < HW) {
    float best = d[tid][0]; int bi = 0;
    for (int n = 1; n < HW; ++n) { float v = d[tid][n]; if (v < best) { best = v; bi = n; } }
    idx1[b * HW + tid] = bi;
  }
  if (tid >= 64 && tid < 64 + HW) {
    int n = tid - 64;
    float best = d[0][n]; int bi = 0;
    for (int m = 1; m < HW; ++m) { float v = d[m][n]; if (v < best) { best = v; bi = m; } }
    idx2[b * HW + n] = bi;
  }
}

// ---------------- branch-2 matching: location NN + top-k by NN distance ----------------
__global__ void match2_k(const float* __restrict__ loc,
                         int* __restrict__ selA, int* __restrict__ nnA,
                         int* __restrict__ selB, int* __restrict__ nnB) {
  __shared__ float dl[HW][HW];
  __shared__ float valA[HW], valB[HW];
  __shared__ int idxA[HW], idxB[HW], rA[HW], rB[HW];
  int b = blockIdx.x, tid = threadIdx.x;
  const float* L = loc + (size_t)b * HW * 2;
  for (int pair = tid; pair < HW * HW; pair += 64) {
    int m = pair / HW, n = pair % HW;
    float gx = ((float)(m / 7) + 0.5f) * 32.f;
    float gy = ((float)(m % 7) + 0.5f) * 32.f;
    float lx = L[n * 2], ly = L[n * 2 + 1];
    float a2 = gx * gx + gy * gy, b2 = lx * lx + ly * ly;
    dl[m][n] = a2 + b2 - 2.f * (gx * lx + gy * ly);  // same algebraic form as reference
  }
  __syncthreads();
  if (tid < HW) {
    float best = dl[tid][0]; int bi = 0;
    for (int n = 1; n < HW; ++n) { float v = dl[tid][n]; if (v < best) { best = v; bi = n; } }
    valA[tid] = best; idxA[tid] = bi;
    best = dl[0][tid]; bi = 0;
    for (int m = 1; m < HW; ++m) { float v = dl[m][tid]; if (v < best) { best = v; bi = m; } }
    valB[tid] = best; idxB[tid] = bi;
  }
  __syncthreads();
  if (tid < HW) {
    float v = valA[tid]; int r = 0;
    for (int m = 0; m < HW; ++m) r += (valA[m] < v) || (valA[m] == v && m < tid);
    rA[tid] = r;
    v = valB[tid]; r = 0;
    for (int m = 0; m < HW; ++m) r += (valB[m] < v) || (valB[m] == v && m < tid);
    rB[tid] = r;
  }
  __syncthreads();
  if (tid == 0) {
    int cnt = 0;
    for (int m = 0; m < HW; ++m) if (rA[m] < 20) { selA[b * 20 + cnt] = m; nnA[b * 20 + cnt] = idxA[m]; ++cnt; }
  }
  if (tid == 1) {
    int cnt = 0;
    for (int m = 0; m < HW; ++m) if (rB[m] < 4) { selB[b * 4 + cnt] = m; nnB[b * 4 + cnt] = idxB[m]; ++cnt; }
  }
}

// ---------------- per-column stats: means + std-term + cov-diagonal term ----------------
__global__ void colstats_k(const float* __restrict__ sx, const int* __restrict__ gx,
                           const float* __restrict__ sy, const int* __restrict__ gy,
                           int M, int C, long bs, int ps,
                           float* __restrict__ meanX, float* __restrict__ meanY,
                           float wStd, float wCov, float* __restrict__ acc) {
  __shared__ float red[256];
  int tid = threadIdx.x;
  int gid = blockIdx.x * 256 + tid;
  float val = 0.f;
  if (gid < M * C) {
    int m = gid / C, c = gid % C;
    float sxa = 0.f, sxx = 0.f, sya = 0.f, syy = 0.f;
    for (int b = 0; b < BATCH; ++b) {
      int px_ = gx ? gx[b * M + m] : m;
      int py_ = gy ? gy[b * M + m] : m;
      float x = sx[(size_t)b * bs + (size_t)px_ * ps + c];
      float y = sy[(size_t)b * bs + (size_t)py_ * ps + c];
      sxa += x; sxx += x * x; sya += y; syy += y * y;
    }
    float mx = sxa * (1.f / BATCH), my = sya * (1.f / BATCH);
    meanX[(size_t)m * C + c] = mx;
    meanY[(size_t)m * C + c] = my;
    float ssx = fmaxf(sxx - (float)BATCH * mx * mx, 0.f);
    float ssy = fmaxf(syy - (float)BATCH * my * my, 0.f);
    float stx = sqrtf(ssx * (1.f / 255.f) + 1e-5f);
    float sty = sqrtf(ssy * (1.f / 255.f) + 1e-5f);
    val = wStd * (fmaxf(1.f - stx, 0.f) + fmaxf(1.f - sty, 0.f))
        - wCov * (ssx * ssx + ssy * ssy);  // subtract diag^2 part of ||cov||^2
  }
  red[tid] = val; __syncthreads();
  for (int s = 128; s > 0; s >>= 1) { if (tid < s) red[tid] += red[tid + s]; __syncthreads(); }
  if (tid == 0) atomicAdd(acc, red[0]);
}

// ---------------- per-batch invariance term ----------------
__global__ void inv_k(const float* __restrict__ sx, const int* __restrict__ gx,
                      const float* __restrict__ sy, const int* __restrict__ gy,
                      int M, int C, long bs, int ps, float wInv,
                      float* __restrict__ batchAcc) {
  __shared__ float red[256];
  int b = blockIdx.x, tid = threadIdx.x;
  float s = 0.f;
  for (int m = 0; m < M; ++m) {
    int px_ = gx ? gx[b * M + m] : m;
    int py_ = gy ? gy[b * M + m] : m;
    const float* X = sx + (size_t)b * bs + (size_t)px_ * ps;
    const float* Y = sy + (size_t)b * bs + (size_t)py_ * ps;
    for (int c = tid; c < C; c += 256) { float d = X[c] - Y[c]; s += d * d; }
  }
  red[tid] = s; __syncthreads();
  for (int t = 128; t > 0; t >>= 1) { if (tid < t) red[tid] += red[tid + t]; __syncthreads(); }
  if (tid == 0) atomicAdd(batchAcc + b, wInv * red[0]);
}

// ---------------- centered Gram Frobenius via fp32 WMMA (16x16x4) ----------------
// ||Xc^T Xc||_F^2 == ||Xc Xc^T||_F^2  (B=256 Gram, 136 upper-triangle 16x16 tiles)
__global__ void gram_k(const float* __restrict__ src, const int* __restrict__ g,
                       int M, int K, long bs, int ps,
                       const float* __restrict__ meanBase,
                       float wCov, float* __restrict__ acc) {
  __shared__ float red[256];
  int tid = threadIdx.x;
  int gw = blockIdx.x * 8 + (tid >> 5);   // one wave32 per tile
  int total = M * 136;
  float partial = 0.f;
  if (gw < total) {
    int m = gw / 136;
    int t = gw % 136;
    int i = 0, rem = t;
    while (rem >= 16 - i) { rem -= 16 - i; ++i; }
    int j = i + rem;                       // upper-triangle tile (i<=j)
    int lane = tid & 31;
    int r  = lane & 15;
    int kh = (lane >> 4) << 1;             // lanes 0-15: K 0,1 ; lanes 16-31: K 2,3
    int rowA = i * 16 + r, rowB = j * 16 + r;
    int posA = g ? g[rowA * M + m] : m;
    int posB = g ? g[rowB * M + m] : m;
    const float2* pA = (const float2*)(src + (size_t)rowA * bs + (size_t)posA * ps);
    const float2* pB = (const float2*)(src + (size_t)rowB * bs + (size_t)posB * ps);
    const float2* mu = (const float2*)(meanBase + (size_t)m * K);
    v8f accv = {0.f, 0.f, 0.f, 0.f, 0.f, 0.f, 0.f, 0.f};
    for (int k = 0; k < K; k += 4) {
      int h = (k + kh) >> 1;
      float2 av = pA[h], bv = pB[h], mv = mu[h];
      v2f a;  a.x  = av.x - mv.x; a.y  = av.y - mv.y;
      v2f bb; bb.x = bv.x - mv.x; bb.y = bv.y - mv.y;
      accv = __builtin_amdgcn_wmma_f32_16x16x4_f32(false, a, false, bb,
                                                   (short)0, accv, false, false);
    }
    float s = 0.f;
#pragma unroll
    for (int e = 0; e < 8; ++e) s += accv[e] * accv[e];
    partial = (i == j) ? s : 2.f * s;      // symmetric off-diagonal tile counts twice
  }
  red[tid] = partial; __syncthreads();
  for (int s2 = 128; s2 > 0; s2 >>= 1) { if (tid < s2) red[tid] += red[tid + s2]; __syncthreads(); }
  if (tid == 0) atomicAdd(acc, wCov * red[0]);
}

// ---------------- combine ----------------
__global__ void final_k(const float* __restrict__ batchAcc,
                        const float* __restrict__ scalarAcc,
                        float* __restrict__ out) {
  int b = threadIdx.x;
  out[b] = batchAcc[b] + scalarAcc[0];
}

extern "C" void kernel_launch(void* const* d_in, const int* in_sizes, int n_in,
                              void* d_out, int out_size, void* d_ws, size_t ws_size,
                              hipStream_t stream) {
  const float* maps1 = (const float*)d_in[0];
  const float* maps2 = (const float*)d_in[1];
  const float* px    = (const float*)d_in[2];
  const float* py    = (const float*)d_in[3];
  const float* loc   = (const float*)d_in[4];
  float* out = (float*)d_out;

  float* ws = (float*)d_ws;
  const size_t ST = (size_t)BATCH * HW * CH;
  float* t1   = ws;
  float* t2   = ws + ST;
  float* mean = ws + 2 * ST;
  float *mP0x = mean,           *mP0y = mean + 25088;
  float *mP1x = mean + 50176,   *mP1y = mean + 75264;
  float *mP2x = mean + 100352,  *mP2y = mean + 110592;
  float *mP3x = mean + 120832,  *mP3y = mean + 122880;
  float* mEx = mean + 124928;
  float* mEy = mEx + EMBD;
  float* batchAcc  = mEy + EMBD;
  float* scalarAcc = batchAcc + 256;
  int* ints = (int*)(scalarAcc + 8);
  int *idx1 = ints,         *idx2 = ints + 12544;
  int *selA = ints + 25088, *nnA  = ints + 30208;
  int *selB = ints + 35328, *nnB  = ints + 36352;

  const long bsM = (long)HW * CH; const int psM = CH;
  const long bsE = EMBD;          const int psE = 0;

  // fold ALPHA=0.5, per-branch 1/2, INV_C=STD_C=25, COV_C=1, 1/(B-1)^2 into weights
  float wStd[4], wCov[4], wInv[4];
  int Mp[4] = {49, 49, 20, 4};
  for (int p = 0; p < 4; ++p) {
    float fm = (float)Mp[p];
    wInv[p] = 0.25f * 25.f / (fm * 512.f);
    wStd[p] = 0.25f * 25.f / (fm * 512.f * 2.f);
    wCov[p] = 0.25f / (fm * 512.f * 2.f * 255.f * 255.f);
  }
  float wInvE = 0.5f * 25.f / 8192.f;
  float wStdE = 0.5f * 25.f / (8192.f * 2.f);
  float wCovE = 0.5f / (8192.f * 255.f * 255.f);

  zero_k<<<2, 256, 0, stream>>>(batchAcc, 264);
  transpose_k<<<BATCH * 8, 256, 0, stream>>>(maps1, t1);
  transpose_k<<<BATCH * 8, 256, 0, stream>>>(maps2, t2);
  match1_k<<<BATCH, 256, 0, stream>>>(t1, t2, idx1, idx2);
  match2_k<<<BATCH, 64, 0, stream>>>(loc, selA, nnA, selB, nnB);

  auto csB = [](int M, int C) { return (M * C + 255) / 256; };
  colstats_k<<<csB(49,512), 256, 0, stream>>>(t1, nullptr, t2, idx1, 49, 512, bsM, psM, mP0x, mP0y, wStd[0], wCov[0], scalarAcc);
  colstats_k<<<csB(49,512), 256, 0, stream>>>(t2, nullptr, t1, idx2, 49, 512, bsM, psM, mP1x, mP1y, wStd[1], wCov[1], scalarAcc);
  colstats_k<<<csB(20,512), 256, 0, stream>>>(t1, selA,    t2, nnA,  20, 512, bsM, psM, mP2x, mP2y, wStd[2], wCov[2], scalarAcc);
  colstats_k<<<csB(4,512),  256, 0, stream>>>(t2, selB,    t1, nnB,   4, 512, bsM, psM, mP3x, mP3y, wStd[3], wCov[3], scalarAcc);
  colstats_k<<<csB(1,8192), 256, 0, stream>>>(px, nullptr, py, nullptr, 1, 8192, bsE, psE, mEx, mEy, wStdE, wCovE, scalarAcc);

  inv_k<<<BATCH, 256, 0, stream>>>(t1, nullptr, t2, idx1, 49, 512, bsM, psM, wInv[0], batchAcc);
  inv_k<<<BATCH, 256, 0, stream>>>(t2, nullptr, t1, idx2, 49, 512, bsM, psM, wInv[1], batchAcc);
  inv_k<<<BATCH, 256, 0, stream>>>(t1, selA,    t2, nnA,  20, 512, bsM, psM, wInv[2], batchAcc);
  inv_k<<<BATCH, 256, 0, stream>>>(t2, selB,    t1, nnB,   4, 512, bsM, psM, wInv[3], batchAcc);
  inv_k<<<BATCH, 256, 0, stream>>>(px, nullptr, py, nullptr, 1, 8192, bsE, psE, wInvE, batchAcc);

  auto grB = [](int M) { return (M * 136 + 7) / 8; };
  gram_k<<<grB(49), 256, 0, stream>>>(t1, nullptr, 49, 512, bsM, psM, mP0x, wCov[0], scalarAcc);
  gram_k<<<grB(49), 256, 0, stream>>>(t2, idx1,    49, 512, bsM, psM, mP0y, wCov[0], scalarAcc);
  gram_k<<<grB(49), 256, 0, stream>>>(t2, nullptr, 49, 512, bsM, psM, mP1x, wCov[1], scalarAcc);
  gram_k<<<grB(49), 256, 0, stream>>>(t1, idx2,    49, 512, bsM, psM, mP1y, wCov[1], scalarAcc);
  gram_k<<<grB(20), 256, 0, stream>>>(t1, selA,    20, 512, bsM, psM, mP2x, wCov[2], scalarAcc);
  gram_k<<<grB(20), 256, 0, stream>>>(t2, nnA,     20, 512, bsM, psM, mP2y, wCov[2], scalarAcc);
  gram_k<<<grB(4),  256, 0, stream>>>(t2, selB,     4, 512, bsM, psM, mP3x, wCov[3], scalarAcc);
  gram_k<<<grB(4),  256, 0, stream>>>(t1, nnB,      4, 512, bsM, psM, mP3y, wCov[3], scalarAcc);
  gram_k<<<grB(1),  256, 0, stream>>>(px, nullptr,  1, 8192, bsE, psE, mEx, wCovE, scalarAcc);
  gram_k<<<grB(1),  256, 0, stream>>>(py, nullptr,  1, 8192, bsE, psE, mEy, wCovE, scalarAcc);

  final_k<<<1, 256, 0, stream>>>(batchAcc, scalarAcc, out);
}